// opool_54606214202265
// MI455X (gfx1250) — compile-verified
//
#include <hip/hip_runtime.h>
#include <hip/hip_bf16.h>

// Orientation-pooling argmax + slice gather.
// x: [B, 12*C, H, W] fp32 -> out: [B, C, H, W] fp32.
// Memory-bound: ~470 MB of traffic, roofline ~20us at 23.3 TB/s.

#define HW      16384                 // 128*128
#define DORIENT 12
#define NSCAN   (DORIENT * HW)        // 196608 elements per (b,c) group
#define N4      (NSCAN / 4)           // 49152 float4 per group
#define COPY4   (HW / 4)              // 4096 float4 per output slice
#define THREADS 1024                  // 32 wave32 waves per block
#define NWAVES  (THREADS / 32)
#define CPT     (COPY4 / THREADS)     // 4 float4 copied per thread

// ---- gfx1250 async global<->LDS data movers (ASYNCcnt), with fallback ----
#if __has_builtin(__builtin_amdgcn_global_load_async_to_lds_b128) && \
    __has_builtin(__builtin_amdgcn_global_store_async_from_lds_b128)
#define USE_ASYNC_LDS 1
#else
#define USE_ASYNC_LDS 0
#endif

#if __has_builtin(__builtin_amdgcn_s_wait_asynccnt)
#define WAIT_ASYNC0() __builtin_amdgcn_s_wait_asynccnt(0)
#else
#define WAIT_ASYNC0() asm volatile("s_wait_asynccnt 0" ::: "memory")
#endif

// Builtin signature (verified via hipcc diagnostic; HIP prints AS1 as `__device__`):
//   void __builtin_amdgcn_global_load_async_to_lds_b128(
//       int4 addrspace(1)* gaddr, int4 addrspace(3)* lds, imm offset, imm cpol)
typedef int v4i __attribute__((vector_size(16)));
typedef __attribute__((address_space(1))) v4i* gptr_v4i;   // global
typedef __attribute__((address_space(3))) v4i* lptr_v4i;   // LDS

__global__ __launch_bounds__(THREADS) void opool_argmax_gather(
    const float* __restrict__ x, float* __restrict__ out)
{
    const unsigned bc  = blockIdx.x;     // bc = b*C + c indexes both in & out
    const unsigned tid = threadIdx.x;

    const float* __restrict__ grp = x + (size_t)bc * NSCAN;
    const float4* __restrict__ g4 = (const float4*)grp;

    // ---------------- Phase 1: argmax over 12*H*W ----------------------------
    // Branch-free: strict '>' keeps the first (lowest-index) max within a
    // thread because each thread scans indices in increasing order. The
    // cross-lane/cross-wave tie-break (lowest flat index) happens in the
    // reductions below, matching jnp.argmax first-occurrence semantics.
    float    best  = -__builtin_huge_valf();
    unsigned besti = 0u;

#pragma unroll 8
    for (unsigned j = tid; j < N4; j += THREADS) {   // 48 B128 loads / thread
        float4 v = g4[j];
        unsigned e = 4u * j;
        bool c0 = v.x > best;  best = c0 ? v.x : best;  besti = c0 ? e      : besti;
        bool c1 = v.y > best;  best = c1 ? v.y : best;  besti = c1 ? e + 1u : besti;
        bool c2 = v.z > best;  best = c2 ? v.z : best;  besti = c2 ? e + 2u : besti;
        bool c3 = v.w > best;  best = c3 ? v.w : best;  besti = c3 ? e + 3u : besti;
    }

    // wave32 butterfly reduction (lowest flat index wins ties)
#pragma unroll
    for (int off = 16; off > 0; off >>= 1) {
        float    ov = __shfl_xor(best, off, 32);
        unsigned oi = (unsigned)__shfl_xor((int)besti, off, 32);
        bool take = (ov > best) | ((ov == best) & (oi < besti));
        best  = take ? ov : best;
        besti = take ? oi : besti;
    }

    __shared__ float    s_val[NWAVES];
    __shared__ unsigned s_idx[NWAVES];
    __shared__ unsigned s_win;

    if ((tid & 31u) == 0u) { s_val[tid >> 5] = best; s_idx[tid >> 5] = besti; }
    __syncthreads();

    if (tid == 0) {
        float    b0 = s_val[0];
        unsigned i0 = s_idx[0];
#pragma unroll
        for (int w = 1; w < NWAVES; ++w) {
            float ov = s_val[w]; unsigned oi = s_idx[w];
            bool take = (ov > b0) | ((ov == b0) & (oi < i0));
            b0 = take ? ov : b0;
            i0 = take ? oi : i0;
        }
        s_win = i0 / HW;                       // winning orientation d in [0,12)
    }
    __syncthreads();
    const unsigned d = s_win;

    // ---------------- Phase 2: gather the winning 128x128 slice --------------
    const float4* __restrict__ w4 = (const float4*)(grp + (size_t)d * HW);
    float4* __restrict__       o4 = (float4*)(out + (size_t)bc * HW);

#if USE_ASYNC_LDS
    // gfx1250 async path: stream global -> LDS -> global with the async data
    // movers (ASYNCcnt). Each wave only stores LDS regions it loaded itself,
    // so the per-wave s_wait_asynccnt between the load group and the store
    // group is the only synchronization required.
    __shared__ float4 stage[COPY4];            // 64 KB staging (<= 320 KB/WGP)
#pragma unroll
    for (unsigned k = 0; k < CPT; ++k) {
        unsigned j = k * THREADS + tid;
        __builtin_amdgcn_global_load_async_to_lds_b128(
            (gptr_v4i)(w4 + j), (lptr_v4i)&stage[j], 0, 0);
    }
    WAIT_ASYNC0();
#pragma unroll
    for (unsigned k = 0; k < CPT; ++k) {
        unsigned j = k * THREADS + tid;
        __builtin_amdgcn_global_store_async_from_lds_b128(
            (gptr_v4i)(o4 + j), (lptr_v4i)&stage[j], 0, 0);
    }
    WAIT_ASYNC0();
#else
    // Fallback: direct B128 copy.
#pragma unroll
    for (unsigned k = 0; k < CPT; ++k) {
        unsigned j = k * THREADS + tid;
        o4[j] = w4[j];
    }
#endif
}

extern "C" void kernel_launch(void* const* d_in, const int* in_sizes, int n_in,
                              void* d_out, int out_size, void* d_ws, size_t ws_size,
                              hipStream_t stream) {
    (void)n_in; (void)d_ws; (void)ws_size; (void)out_size;
    const float* x   = (const float*)d_in[0];
    float*       out = (float*)d_out;
    const int nbc = in_sizes[0] / NSCAN;       // B*C = 512 groups
    opool_argmax_gather<<<dim3(nbc), dim3(THREADS), 0, stream>>>(x, out);
}